// UnrolledSolver_19628000543342
// MI455X (gfx1250) — compile-verified
//
#include <hip/hip_runtime.h>
#include <math.h>

// Problem constants (from reference)
#define N_BUS 2000
#define N2    4000          // 2N unknowns
#define NP    4096          // padded matrix dim (identity padding rows/cols)
#define LDA   4104          // leading dim (avoid power-of-2 cache set conflicts)
#define NB    64            // LU block size
#define NBLK  (NP / NB)     // 64 block steps
#define NEDGE 6000
#define T_IT  5
#define EPSR  1e-6f

typedef __attribute__((ext_vector_type(2))) float v2f;
typedef __attribute__((ext_vector_type(8))) float v8f;

// ---------------------------------------------------------------- clear / init
__global__ void k_clear_J(float* __restrict__ J) {
  size_t i = (size_t)blockIdx.x * blockDim.x + threadIdx.x;
  size_t n4 = (size_t)NP * LDA / 4;
  float4* p = (float4*)J;
  if (i < n4) p[i] = make_float4(0.f, 0.f, 0.f, 0.f);
}

__global__ void k_clear_vec(float* __restrict__ F, float* __restrict__ Pa,
                            float* __restrict__ Qa) {
  int i = blockIdx.x * blockDim.x + threadIdx.x;
  if (i < NP) F[i] = 0.f;
  if (i < 2048) { Pa[i] = 0.f; Qa[i] = 0.f; }
}

__global__ void k_init_x(float* __restrict__ xc, const float* __restrict__ x_in) {
  int i = blockIdx.x * blockDim.x + threadIdx.x;
  if (i < N2) xc[i] = x_in[i];
}

__global__ void k_write_out(float* __restrict__ out, const float* __restrict__ xc) {
  int i = blockIdx.x * blockDim.x + threadIdx.x;
  if (i < N2) out[i] = xc[i];
}

// ---------------------------------------------------------------- edge scatter
// Builds branch-flow sums and scatters the 16 Jacobian entries per edge.
// keep-mask (row zeroing for slack/PV rows) is folded in at scatter time.
__global__ void k_edges(const float* __restrict__ xc, const int* __restrict__ ei,
                        const float* __restrict__ br_r, const float* __restrict__ br_x,
                        const float* __restrict__ g_fr, const float* __restrict__ b_fr,
                        const float* __restrict__ g_to, const float* __restrict__ b_to,
                        const float* __restrict__ tap, const float* __restrict__ shift,
                        const int* __restrict__ bt,
                        float* __restrict__ J, float* __restrict__ Pa,
                        float* __restrict__ Qa) {
  int e = blockIdx.x * blockDim.x + threadIdx.x;
  if (e >= NEDGE) return;
  int s = ei[e], d = ei[NEDGE + e];
  float va_s = xc[s], va_d = xc[d];
  float v_i = xc[N_BUS + s], v_j = xc[N_BUS + d];
  float r = br_r[e], xx = br_x[e];
  float den = r * r + xx * xx;
  float gS = r / den, bS = -xx / den;
  float tp = tap[e], sh = shift[e];
  float th = va_s - va_d;
  float sf, cf;
  sincosf(th - sh, &sf, &cf);
  // cos(shift-th)=cf, sin(shift-th)=-sf (exact identity)
  float ct = cf, st = -sf;
  float vij = v_i * v_j / tp;
  float gsum_f = gS + g_fr[e], bsum_f = bS + b_fr[e];
  float gsum_t = gS + g_to[e], bsum_t = bS + b_to[e];
  float vit = v_i / tp;

  float Pf = vit * vit * gsum_f + vij * (-gS * cf - bS * sf);
  float Qf = -vit * vit * bsum_f + vij * (-gS * sf + bS * cf);
  float Pt = v_j * v_j * gsum_t + vij * (-gS * ct - bS * st);
  float Qt = -v_j * v_j * bsum_t + vij * (-gS * st + bS * ct);
  atomicAdd(&Pa[s], Pf); atomicAdd(&Pa[d], Pt);
  atomicAdd(&Qa[s], Qf); atomicAdd(&Qa[d], Qt);

  float Af = gS * cf + bS * sf, Bf = gS * sf - bS * cf;
  float At = gS * ct + bS * st, Bt = gS * st - bS * ct;

  int bts = bt[s], btd = bt[d];
  float kPs = (bts == 3) ? 0.f : 1.f, kPd = (btd == 3) ? 0.f : 1.f;
  float kQs = (bts >= 2) ? 0.f : 1.f, kQd = (btd >= 2) ? 0.f : 1.f;

  float dPfs_th_s = vij * Bf, dPfs_th_d = -vij * Bf;
  float dPfs_vm_s = 2.f * v_i / (tp * tp) * gsum_f - (v_j / tp) * Af;
  float dPfs_vm_d = -(v_i / tp) * Af;
  float dQfs_th_s = -vij * Af, dQfs_th_d = vij * Af;
  float dQfs_vm_s = -2.f * v_i / (tp * tp) * bsum_f - (v_j / tp) * Bf;
  float dQfs_vm_d = -(v_i / tp) * Bf;
  float dPfd_th_d = vij * Bt, dPfd_th_s = -vij * Bt;
  float dPfd_vm_d = 2.f * v_j * gsum_t - (v_i / tp) * At;
  float dPfd_vm_s = -(v_j / tp) * At;
  float dQfd_th_d = -vij * At, dQfd_th_s = vij * At;
  float dQfd_vm_d = -2.f * v_j * bsum_t - (v_i / tp) * Bt;
  float dQfd_vm_s = -(v_j / tp) * Bt;

  size_t Ls = (size_t)LDA;
  // P row of s
  atomicAdd(&J[(size_t)s * Ls + s],          kPs * -dPfs_th_s);
  atomicAdd(&J[(size_t)s * Ls + d],          kPs * -dPfs_th_d);
  atomicAdd(&J[(size_t)s * Ls + N_BUS + s],  kPs * -dPfs_vm_s);
  atomicAdd(&J[(size_t)s * Ls + N_BUS + d],  kPs * -dPfs_vm_d);
  // P row of d
  atomicAdd(&J[(size_t)d * Ls + d],          kPd * -dPfd_th_d);
  atomicAdd(&J[(size_t)d * Ls + s],          kPd * -dPfd_th_s);
  atomicAdd(&J[(size_t)d * Ls + N_BUS + d],  kPd * -dPfd_vm_d);
  atomicAdd(&J[(size_t)d * Ls + N_BUS + s],  kPd * -dPfd_vm_s);
  // Q row of s
  atomicAdd(&J[(size_t)(N_BUS + s) * Ls + s],         kQs * -dQfs_th_s);
  atomicAdd(&J[(size_t)(N_BUS + s) * Ls + d],         kQs * -dQfs_th_d);
  atomicAdd(&J[(size_t)(N_BUS + s) * Ls + N_BUS + s], kQs * -dQfs_vm_s);
  atomicAdd(&J[(size_t)(N_BUS + s) * Ls + N_BUS + d], kQs * -dQfs_vm_d);
  // Q row of d
  atomicAdd(&J[(size_t)(N_BUS + d) * Ls + d],         kQd * -dQfd_th_d);
  atomicAdd(&J[(size_t)(N_BUS + d) * Ls + s],         kQd * -dQfd_th_s);
  atomicAdd(&J[(size_t)(N_BUS + d) * Ls + N_BUS + d], kQd * -dQfd_vm_d);
  atomicAdd(&J[(size_t)(N_BUS + d) * Ls + N_BUS + s], kQd * -dQfd_vm_s);
}

// -------------------------------------------- diagonal terms + residual vector
__global__ void k_diag_resid(const float* __restrict__ xc,
                             const float* __restrict__ p_spec,
                             const float* __restrict__ q_spec,
                             const float* __restrict__ gs, const float* __restrict__ bs,
                             const int* __restrict__ bt,
                             const float* __restrict__ vmset,
                             const float* __restrict__ Pa, const float* __restrict__ Qa,
                             float* __restrict__ J, float* __restrict__ F) {
  int i = blockIdx.x * blockDim.x + threadIdx.x;
  size_t Ls = (size_t)LDA;
  if (i < N_BUS) {
    float vm = xc[N_BUS + i], va = xc[i];
    int b = bt[i];
    bool sl = (b == 3), pvsl = (b >= 2);
    float kP = sl ? 0.f : 1.f, kQ = pvsl ? 0.f : 1.f;
    float g = gs[i], bsh = bs[i];
    atomicAdd(&J[(size_t)i * Ls + i], (sl ? 1.f : 0.f) + EPSR);
    atomicAdd(&J[(size_t)i * Ls + N_BUS + i], kP * (-2.f * vm * g));
    atomicAdd(&J[(size_t)(N_BUS + i) * Ls + N_BUS + i],
              kQ * (2.f * vm * bsh) + (pvsl ? 1.f : 0.f) + EPSR);
    float Pc = Pa[i] + vm * vm * g;
    float Qc = Qa[i] - vm * vm * bsh;
    F[i]          = sl   ? va               : (p_spec[i] - Pc);
    F[N_BUS + i]  = pvsl ? (vm - vmset[i])  : (q_spec[i] - Qc);
  } else if (i >= N2 && i < NP) {
    J[(size_t)i * Ls + i] = 1.0f;   // identity padding
    F[i] = 0.f;
  }
}

// ---------------------------------------------------------------- LU kernels
__global__ void k_lu_diag(float* __restrict__ J, int kb) {
  __shared__ float T[NB][NB + 1];
  int t = threadIdx.x;                       // 0..63
  size_t base = (size_t)kb * NB;
  for (int r = 0; r < NB; ++r) T[r][t] = J[(base + r) * LDA + base + t];
  for (int p = 0; p < NB; ++p) {
    __syncthreads();
    float inv = 1.0f / T[p][p];
    if (t > p) {
      float l = T[t][p] * inv;
      T[t][p] = l;
      for (int c = p + 1; c < NB; ++c) T[t][c] -= l * T[p][c];
    }
  }
  __syncthreads();
  for (int r = 0; r < NB; ++r) J[(base + r) * LDA + base + t] = T[r][t];
}

// L21 = A21 * U11^{-1}  (one workgroup per 64-row tile below the pivot block)
__global__ void k_lu_trsm_col(float* __restrict__ J, int kb) {
  __shared__ float U[NB][NB + 1];
  __shared__ float A[NB][NB + 1];
  int t = threadIdx.x;
  size_t kbase = (size_t)kb * NB;
  size_t rbase = kbase + NB + (size_t)blockIdx.x * NB;
  for (int r = 0; r < NB; ++r) {
    U[r][t] = J[(kbase + r) * LDA + kbase + t];
    A[r][t] = J[(rbase + r) * LDA + kbase + t];
  }
  __syncthreads();
  for (int j = 0; j < NB; ++j) {            // thread t owns row t
    float a = A[t][j];
    for (int p = 0; p < j; ++p) a -= A[t][p] * U[p][j];
    A[t][j] = a / U[j][j];
  }
  __syncthreads();
  for (int r = 0; r < NB; ++r) J[(rbase + r) * LDA + kbase + t] = A[r][t];
}

// U12 = L11^{-1} * A12  (one workgroup per 64-col tile right of the pivot block)
__global__ void k_lu_trsm_row(float* __restrict__ J, int kb) {
  __shared__ float L[NB][NB + 1];
  __shared__ float A[NB][NB + 1];
  int t = threadIdx.x;
  size_t kbase = (size_t)kb * NB;
  size_t cbase = kbase + NB + (size_t)blockIdx.x * NB;
  for (int r = 0; r < NB; ++r) {
    L[r][t] = J[(kbase + r) * LDA + kbase + t];
    A[r][t] = J[(kbase + r) * LDA + cbase + t];
  }
  __syncthreads();
  for (int i = 1; i < NB; ++i) {            // thread t owns column t; L unit-lower
    float a = A[i][t];
    for (int p = 0; p < i; ++p) a -= L[i][p] * A[p][t];
    A[i][t] = a;
  }
  __syncthreads();
  for (int r = 0; r < NB; ++r) J[(kbase + r) * LDA + cbase + t] = A[r][t];
}

// Trailing update: C -= L21 * U12, 64x64x64 tile per workgroup via f32 WMMA.
// f32 WMMA has no A-negation (NEG is CNeg-only), so we stage -L21 in LDS.
__global__ void __launch_bounds__(256) k_lu_update(float* __restrict__ J, int kb) {
  __shared__ float As[NB][NB + 4];   // stride 68: conflict-free A-frag gathers
  __shared__ float Bs[NB][NB + 8];   // stride 72: conflict-free B-frag gathers
  int tid = threadIdx.x;
  size_t kbase = (size_t)kb * NB;
  size_t rbase = kbase + NB + (size_t)blockIdx.x * NB;   // C row tile origin
  size_t cbase = kbase + NB + (size_t)blockIdx.y * NB;   // C col tile origin
  for (int idx = tid; idx < NB * NB; idx += 256) {
    int r = idx >> 6, c = idx & 63;
    As[r][c] = -J[(rbase + r) * LDA + kbase + c];        // negated L21
    Bs[r][c] =  J[(kbase + r) * LDA + cbase + c];        // U12
  }
  __syncthreads();

  int wave = tid >> 5, lane = tid & 31;
  int half = lane >> 4, n = lane & 15;
  // Each wave owns two adjacent 16x16 C subtiles (same row band).
  int st0 = wave * 2;
  int si  = st0 >> 2;
  int sj0 = st0 & 3, sj1 = sj0 + 1;
  size_t crow  = rbase + (size_t)si * 16;
  size_t ccol0 = cbase + (size_t)sj0 * 16;
  size_t ccol1 = cbase + (size_t)sj1 * 16;

  v8f c0, c1;
#pragma unroll
  for (int v = 0; v < 8; ++v) {
    size_t rr = (crow + v + 8 * half) * LDA;             // C/D layout: VGPR v -> M=v(+8)
    c0[v] = J[rr + ccol0 + n];
    c1[v] = J[rr + ccol1 + n];
  }

  int am = si * 16 + n;                                  // A-frag: M = lane&15
#pragma unroll
  for (int kk = 0; kk < NB; kk += 4) {
    v2f a, b0, b1;
    a[0]  = As[am][kk + 2 * half];                       // K = 0/2 per lane half
    a[1]  = As[am][kk + 2 * half + 1];                   // K = 1/3
    b0[0] = Bs[kk + 2 * half][sj0 * 16 + n];
    b0[1] = Bs[kk + 2 * half + 1][sj0 * 16 + n];
    b1[0] = Bs[kk + 2 * half][sj1 * 16 + n];
    b1[1] = Bs[kk + 2 * half + 1][sj1 * 16 + n];
    c0 = __builtin_amdgcn_wmma_f32_16x16x4_f32(false, a, false, b0, (short)0, c0,
                                               false, false);
    c1 = __builtin_amdgcn_wmma_f32_16x16x4_f32(false, a, false, b1, (short)0, c1,
                                               false, false);
  }

#pragma unroll
  for (int v = 0; v < 8; ++v) {
    size_t rr = (crow + v + 8 * half) * LDA;
    J[rr + ccol0 + n] = c0[v];
    J[rr + ccol1 + n] = c1[v];
  }
}

// ----------------------------------------------------- blocked triangular solves
__global__ void __launch_bounds__(256) k_trsv_lower(const float* __restrict__ J,
                                                    float* __restrict__ F) {
  __shared__ float Lb[NB][NB + 1];
  __shared__ float yb[NB];
  int t = threadIdx.x;
  for (int kb = 0; kb < NBLK; ++kb) {
    size_t base = (size_t)kb * NB;
    for (int idx = t; idx < NB * NB; idx += 256) {
      int r = idx >> 6, c = idx & 63;
      Lb[r][c] = J[(base + r) * LDA + base + c];
    }
    if (t < NB) yb[t] = F[base + t];
    __syncthreads();
    for (int j = 0; j < NB; ++j) {          // unit-lower forward substitution
      if (t < NB && t > j) yb[t] -= Lb[t][j] * yb[j];
      __syncthreads();
    }
    if (t < NB) F[base + t] = yb[t];
    __syncthreads();
    for (size_t r = base + NB + t; r < NP; r += 256) {
      float acc = 0.f;
#pragma unroll 8
      for (int j = 0; j < NB; ++j) acc += J[r * LDA + base + j] * yb[j];
      F[r] -= acc;
    }
    __syncthreads();
  }
}

__global__ void __launch_bounds__(256) k_trsv_upper(const float* __restrict__ J,
                                                    float* __restrict__ F) {
  __shared__ float Ub[NB][NB + 1];
  __shared__ float yb[NB];
  int t = threadIdx.x;
  for (int kb = NBLK - 1; kb >= 0; --kb) {
    size_t base = (size_t)kb * NB;
    for (int idx = t; idx < NB * NB; idx += 256) {
      int r = idx >> 6, c = idx & 63;
      Ub[r][c] = J[(base + r) * LDA + base + c];
    }
    if (t < NB) yb[t] = F[base + t];
    __syncthreads();
    for (int j = NB - 1; j >= 0; --j) {     // back substitution
      if (t == j) yb[j] /= Ub[j][j];
      __syncthreads();
      if (t < j) yb[t] -= Ub[t][j] * yb[j];
      __syncthreads();
    }
    if (t < NB) F[base + t] = yb[t];
    __syncthreads();
    for (size_t r = t; r < base; r += 256) {
      float acc = 0.f;
#pragma unroll 8
      for (int j = 0; j < NB; ++j) acc += J[r * LDA + base + j] * yb[j];
      F[r] -= acc;
    }
    __syncthreads();
  }
}

__global__ void k_update_x(float* __restrict__ xc, const float* __restrict__ F) {
  int i = blockIdx.x * blockDim.x + threadIdx.x;
  if (i < N2) {
    float v = xc[i] - F[i];                  // F holds dx after the two trsv passes
    if (i >= N_BUS) v = fminf(fmaxf(v, 0.5f), 1.5f);
    xc[i] = v;
  }
}

// ---------------------------------------------------------------- launch driver
extern "C" void kernel_launch(void* const* d_in, const int* in_sizes, int n_in,
                              void* d_out, int out_size, void* d_ws, size_t ws_size,
                              hipStream_t stream) {
  (void)in_sizes; (void)n_in; (void)out_size; (void)ws_size;
  const float* x_in      = (const float*)d_in[0];
  const int*   ei        = (const int*)  d_in[1];
  const float* br_r      = (const float*)d_in[2];
  const float* br_x      = (const float*)d_in[3];
  const float* g_fr      = (const float*)d_in[4];
  const float* b_fr      = (const float*)d_in[5];
  const float* g_to      = (const float*)d_in[6];
  const float* b_to      = (const float*)d_in[7];
  const float* tap       = (const float*)d_in[8];
  const float* shift     = (const float*)d_in[9];
  const float* p_spec    = (const float*)d_in[10];
  const float* q_spec    = (const float*)d_in[11];
  const float* gs        = (const float*)d_in[12];
  const float* bs        = (const float*)d_in[13];
  const int*   bus_type  = (const int*)  d_in[14];
  const float* vmset     = (const float*)d_in[15];

  // Workspace layout (floats): J[NP*LDA] | F[NP] | Pa[2048] | Qa[2048] | xc[4096]
  float* J  = (float*)d_ws;
  float* F  = J + (size_t)NP * LDA;
  float* Pa = F + NP;
  float* Qa = Pa + 2048;
  float* xc = Qa + 2048;

  k_init_x<<<16, 256, 0, stream>>>(xc, x_in);

  for (int it = 0; it < T_IT; ++it) {
    k_clear_J<<<(int)(((size_t)NP * LDA / 4 + 255) / 256), 256, 0, stream>>>(J);
    k_clear_vec<<<16, 256, 0, stream>>>(F, Pa, Qa);
    k_edges<<<(NEDGE + 255) / 256, 256, 0, stream>>>(xc, ei, br_r, br_x, g_fr, b_fr,
                                                     g_to, b_to, tap, shift, bus_type,
                                                     J, Pa, Qa);
    k_diag_resid<<<16, 256, 0, stream>>>(xc, p_spec, q_spec, gs, bs, bus_type, vmset,
                                         Pa, Qa, J, F);
    // Blocked LU (no pivoting; EPS-regularized like the reference)
    for (int kb = 0; kb < NBLK; ++kb) {
      k_lu_diag<<<1, NB, 0, stream>>>(J, kb);
      int rem = NBLK - 1 - kb;
      if (rem > 0) {
        k_lu_trsm_col<<<rem, NB, 0, stream>>>(J, kb);
        k_lu_trsm_row<<<rem, NB, 0, stream>>>(J, kb);
        k_lu_update<<<dim3(rem, rem), 256, 0, stream>>>(J, kb);
      }
    }
    k_trsv_lower<<<1, 256, 0, stream>>>(J, F);
    k_trsv_upper<<<1, 256, 0, stream>>>(J, F);
    k_update_x<<<16, 256, 0, stream>>>(xc, F);
  }

  k_write_out<<<16, 256, 0, stream>>>((float*)d_out, xc);
}